// BaseSigKernel_86011015070032
// MI455X (gfx1250) — compile-verified
//
#include <hip/hip_runtime.h>

// Signature kernel (sigkernel PDE), MI455X / gfx1250, wave32.
//
// Roofline: ~220 MFLOP total, 256 KB input, 4 KB output -> latency/parallelism
// bound, not BW/FLOP bound. Design:
//   * inc = (0.25*dX) . dY^T computed with V_WMMA_F32_16X16X4_F32 (f32 matrix
//     pipe, full reference precision); 64 WMMAs per pair, operands staged in
//     LDS, 1/4 dyadic scale folded into the dX increments (stage 1).
//   * Goursat PDE 126x126 solved by a skewed wavefront ladder inside one
//     wave32: lane t owns 4 columns, one __shfl_up per step carries the
//     left-neighbor column with a 2-deep history; 157 steps, registers only.
//   * One wave (block of 32) per (a,b) pair -> 1024 independent waves,
//     no global sync, no workspace, everything L2/LDS resident.

typedef __attribute__((ext_vector_type(2))) float v2f;
typedef __attribute__((ext_vector_type(8))) float v8f;

#define LPATH 64     // path length
#define DDIM  16     // feature dim
#define NG    64     // padded Gram size (63 real + 1 zero row/col)
#define NGRID 126    // PDE grid = 2*(LPATH-1), DYADIC_ORDER=1

__global__ __launch_bounds__(32)
void sigkernel_pde_wmma(const float* __restrict__ xs,
                        const float* __restrict__ ys,
                        float* __restrict__ out,
                        int By) {
  const int p = blockIdx.x;
  const int a = p / By;
  const int b = p - a * By;
  const float* X = xs + (size_t)a * (LPATH * DDIM);
  const float* Y = ys + (size_t)b * (LPATH * DDIM);

  __shared__ float sdX[NG * DDIM];   // 0.25 * path increments of x (row 63 = 0)
  __shared__ float sdY[NG * DDIM];   // path increments of y (row 63 = 0)
  __shared__ float sV[NG * NG];      // scaled double increments (Gram of incs)

  const int t = threadIdx.x;

  // ---- Stage 1: path increments into LDS (finite diff + 1/d^2 fused) ----
  for (int idx = t; idx < NG * DDIM; idx += 32) {
    const int row = idx >> 4;                  // /DDIM
    float dx = 0.f, dy = 0.f;
    if (row < LPATH - 1) {
      dx = 0.25f * (X[idx + DDIM] - X[idx]);   // fold dyadic 1/4 into dX
      dy = Y[idx + DDIM] - Y[idx];
    }
    sdX[idx] = dx;
    sdY[idx] = dy;
  }
  __syncthreads();

  // ---- Stage 2: V = (0.25*dX) * dY^T via V_WMMA_F32_16X16X4_F32 ----
  // A 16x4 f32 frag: lane = (k/2)*16 + m, vgpr = k&1  -> per lane reads
  // sdX[(16I + lane%16)*16 + 4*kc + 2*(lane/16) + {0,1}]; B mirrored.
  const int lo = t & 15;
  const int hi = t >> 4;

  for (int I = 0; I < 4; ++I) {
    v2f afrag[4];
#pragma unroll
    for (int kc = 0; kc < 4; ++kc) {
      const float* ap = &sdX[(16 * I + lo) * DDIM + 4 * kc + 2 * hi];
      afrag[kc][0] = ap[0];
      afrag[kc][1] = ap[1];
    }
    for (int J = 0; J < 4; ++J) {
      v8f acc = {};
#pragma unroll
      for (int kc = 0; kc < 4; ++kc) {
        v2f bfrag;
        const float* bp = &sdY[(16 * J + lo) * DDIM + 4 * kc + 2 * hi];
        bfrag[0] = bp[0];
        bfrag[1] = bp[1];
        // (neg_a, A, neg_b, B, c_mod, C, reuse_a, reuse_b)
        acc = __builtin_amdgcn_wmma_f32_16x16x4_f32(
            false, afrag[kc], false, bfrag, (short)0, acc, false, false);
      }
      // D layout: vgpr v -> row M = 8*hi + v, col N = lo (within tile)
#pragma unroll
      for (int v = 0; v < 8; ++v) {
        sV[(16 * I + 8 * hi + v) * NG + 16 * J + lo] = acc[v];
      }
    }
  }
  __syncthreads();

  // ---- Stage 3: Goursat PDE wavefront ----
  // 1-based K grid (127x127), K[0][*]=K[*][0]=1.
  // Lane t owns columns c = 4t+1 .. 4t+4; at step s it computes row i = s-t.
  // Left-neighbor column values arrive via shfl_up with a 2-deep history.
  // vf[r][jc] = sV[r>>1][jc>>1] (2x2 dyadic repeat); columns >=126 hit the
  // zero-padded Gram col -> vf=0 -> inert cells.
  float kp0 = 1.f, kp1 = 1.f, kp2 = 1.f, kp3 = 1.f;  // K[i-1][c] per column
  float myout  = 1.f;   // my col 4t+4 value of most recently computed row
  float prevRec = 1.f;  // left value received one step ago = K[i-1][4t]

#define PDE_CELL(VV, KP)                                          \
  {                                                               \
    const float vv  = (VV);                                       \
    const float c1  = 1.f + vv * (0.5f + vv * (1.f / 12.f));      \
    const float c2  = 1.f - vv * vv * (1.f / 12.f);               \
    const float old = (KP);                                       \
    const float kn  = (kleft + old) * c1 - kdiag * c2;            \
    kdiag = old; (KP) = kn; kleft = kn;                           \
  }

  for (int s = 1; s <= NGRID + 31; ++s) {
    float rec = __shfl_up(myout, 1, 32);   // lane t-1's last column, row i
    if (t == 0) rec = 1.f;                 // column-0 boundary
    const int i = s - t;
    if (i >= 1 && i <= NGRID) {
      const int r = i - 1;
      const v2f vp = *reinterpret_cast<const v2f*>(&sV[(r >> 1) * NG + 2 * t]);
      float kleft = rec;       // K[i][4t]
      float kdiag = prevRec;   // K[i-1][4t]
      PDE_CELL(vp[0], kp0);
      PDE_CELL(vp[0], kp1);
      PDE_CELL(vp[1], kp2);
      PDE_CELL(vp[1], kp3);
      myout = kp3;
    }
    prevRec = rec;
  }
#undef PDE_CELL

  // K[126][126]: lane 31, column 4*31+2 -> kp1
  if (t == 31) out[p] = kp1;
}

extern "C" void kernel_launch(void* const* d_in, const int* in_sizes, int n_in,
                              void* d_out, int out_size, void* d_ws, size_t ws_size,
                              hipStream_t stream) {
  const float* xs = (const float*)d_in[0];
  const float* ys = (const float*)d_in[1];
  float* out = (float*)d_out;

  const int Bx = in_sizes[0] / (LPATH * DDIM);   // 32
  const int By = in_sizes[1] / (LPATH * DDIM);   // 32

  sigkernel_pde_wmma<<<dim3(Bx * By), dim3(32), 0, stream>>>(xs, ys, out, By);
}